// RelationYOLOX_85916525789351
// MI455X (gfx1250) — compile-verified
//
#include <hip/hip_runtime.h>
#include <cstdint>

// ---------------------------------------------------------------------------
// RelationYOLOX refine: per-(batch,level) exact top-1000 + row gather/concat.
// B=8, levels: hw = 16384/4096/1024, strides 8/16/32.
// Out row (981): [fc 256 | fl 256 | fi 256 | cls 80 | dec 4 | id 128 | obj 1]
// Memory-bound (no matmul): CDNA5 async global<->LDS DMA is the target path.
// ---------------------------------------------------------------------------

#define NB   8
#define NK   1000
#define OUTC 981

// ---- CDNA5 async global<->LDS helpers (inline asm; tracked by ASYNCcnt) ----
__device__ __forceinline__ unsigned lds_off(const void* p) {
  // generic LDS pointer: low 32 bits == workgroup-relative LDS byte address
  return (unsigned)(unsigned long long)(uintptr_t)p;
}
__device__ __forceinline__ void async_load_b32(const float* g, unsigned lds) {
  asm volatile("global_load_async_to_lds_b32 %0, %1, off"
               :: "v"(lds), "v"((unsigned long long)(uintptr_t)g) : "memory");
}
__device__ __forceinline__ void async_load_b128(const float* g, unsigned lds) {
  // requires 16B-aligned global address and 16B-aligned LDS address
  asm volatile("global_load_async_to_lds_b128 %0, %1, off"
               :: "v"(lds), "v"((unsigned long long)(uintptr_t)g) : "memory");
}
__device__ __forceinline__ void async_store_b32_nt(float* g, unsigned lds) {
  // non-temporal: 94 MB streaming output must not evict gather lines from L2
  asm volatile("global_store_async_from_lds_b32 %0, %1, off th:TH_STORE_NT"
               :: "v"((unsigned long long)(uintptr_t)g), "v"(lds) : "memory");
}
__device__ __forceinline__ void wait_async0() {
  asm volatile("s_wait_asynccnt 0" ::: "memory");
}

// ---------------------------------------------------------------------------
// Kernel 1: exact top-1000 per batch for one level.
// Bitonic sort of u64 keys ((ordered_f32<<32) | (0xFFFFFFFF - idx)) in LDS,
// descending -> matches jax.lax.top_k ordering incl. lower-index tiebreak.
// grid = 8 (batches), block = 1024, dynamic LDS = hw * 8 bytes (<= 128 KB,
// fits the 320 KB WGP LDS).
// ---------------------------------------------------------------------------
__global__ void topk_kernel(const float* __restrict__ scores,
                            int* __restrict__ out_idx, int l, int hw) {
  extern __shared__ unsigned long long keys[];
  const int b   = blockIdx.x;
  const int tid = threadIdx.x;
  const float* s = scores + (size_t)b * hw;

  for (int i = tid; i < hw; i += 1024) {
    unsigned u = __float_as_uint(s[i]);
    u = (u & 0x80000000u) ? ~u : (u | 0x80000000u);   // monotonic float->uint
    keys[i] = ((unsigned long long)u << 32) |
              (unsigned long long)(0xFFFFFFFFu - (unsigned)i);
  }
  __syncthreads();

  for (int k = 2; k <= hw; k <<= 1) {
    for (int j = k >> 1; j > 0; j >>= 1) {
      for (int i = tid; i < hw; i += 1024) {
        int ixj = i ^ j;
        if (ixj > i) {
          unsigned long long a = keys[i];
          unsigned long long c = keys[ixj];
          bool desc = ((i & k) == 0);
          if (desc ? (a < c) : (a > c)) { keys[i] = c; keys[ixj] = a; }
        }
      }
      __syncthreads();
    }
  }

  for (int t = tid; t < NK; t += 1024) {
    out_idx[((size_t)b * 3 + l) * NK + t] =
        (int)(0xFFFFFFFFu - (unsigned)keys[t]);
  }
}

// ---------------------------------------------------------------------------
// Kernel 2: assemble one 981-float output row per block (24000 blocks, 256 thr).
// Scattered gathers: global -> LDS via async DMA (b32 for strided channel
// gathers, b128 for aligned contiguous token rows); row leaves LDS -> global
// via non-temporal async stores. s_endpgm's implicit wait-idle drains stores.
// ---------------------------------------------------------------------------
struct Ptrs { const float* p[21]; };

__global__ void gather_kernel(Ptrs P, const int* __restrict__ idxbuf,
                              float* __restrict__ out) {
  __shared__ float row[OUTC];

  const int g = blockIdx.x;            // 0 .. 23999
  const int b = g / 3000;
  const int r = g - b * 3000;
  const int l = r / NK;
  const int t = r - l * NK;
  const int tid = threadIdx.x;         // 256 threads

  const float *cls, *loc, *idp, *obj, *fc, *fl, *fi;
  if (l == 0)      { cls=P.p[ 0]; loc=P.p[ 1]; idp=P.p[ 2]; obj=P.p[ 3];
                     fc =P.p[ 4]; fl =P.p[ 5]; fi =P.p[ 6]; }
  else if (l == 1) { cls=P.p[ 7]; loc=P.p[ 8]; idp=P.p[ 9]; obj=P.p[10];
                     fc =P.p[11]; fl =P.p[12]; fi =P.p[13]; }
  else             { cls=P.p[14]; loc=P.p[15]; idp=P.p[16]; obj=P.p[17];
                     fc =P.p[18]; fl =P.p[19]; fi =P.p[20]; }

  const int hw      = 16384 >> (2 * l);
  const int wshift  = 7 - l;                 // w = 128 >> l
  const float strd  = (float)(8 << l);

  const int p = idxbuf[((size_t)b * 3 + l) * NK + t];

  // --- feature-map transpose-gathers: feat[b, c, p], channel stride = hw ---
  const size_t fbase = ((size_t)b * 256 + tid) * (size_t)hw + (size_t)p;
  async_load_b32(fc + fbase, lds_off(&row[        tid]));
  async_load_b32(fl + fbase, lds_off(&row[256  + tid]));
  async_load_b32(fi + fbase, lds_off(&row[512  + tid]));

  // --- token gathers: contiguous, 16B-aligned rows -> wide b128 DMA ---
  const size_t tok = (size_t)b * hw + (size_t)p;
  if (tid < 20)  // 80 floats = 20 lanes x 16B ; row[768]*4 = 3072 B (16B ok)
    async_load_b128(cls + tok * 80  + tid * 4, lds_off(&row[768 + tid * 4]));
  if (tid < 32)  // 128 floats = 32 lanes x 16B ; row[852]*4 = 3408 B (16B ok)
    async_load_b128(idp + tok * 128 + tid * 4, lds_off(&row[852 + tid * 4]));
  if (tid == 0)
    async_load_b32(obj + tok, lds_off(&row[980]));

  // --- box decode: dec = [lp01*stride + loc_xy, exp(lp23)*stride] ---
  if (tid < 4) {
    float lp = loc[tok * 4 + tid];
    float v;
    if (tid == 0)      v = lp * strd + (float)(p & ((1 << wshift) - 1)) * strd;
    else if (tid == 1) v = lp * strd + (float)(p >> wshift) * strd;
    else               v = expf(lp) * strd;
    row[848 + tid] = v;
  }

  wait_async0();       // all async gathers landed in LDS
  __syncthreads();     // DS store for dec visible too

  // --- write out 981 floats, LDS -> global, coalesced NT async stores ---
  const size_t obase = (((size_t)b * 3 + l) * NK + t) * (size_t)OUTC;
  for (int i = tid; i < OUTC; i += 256)
    async_store_b32_nt(out + obase + i, lds_off(&row[i]));
  // s_endpgm performs implicit wait-idle -> stores drain before wave ends
}

// ---------------------------------------------------------------------------
extern "C" void kernel_launch(void* const* d_in, const int* in_sizes, int n_in,
                              void* d_out, int out_size, void* d_ws, size_t ws_size,
                              hipStream_t stream) {
  (void)in_sizes; (void)n_in; (void)out_size; (void)ws_size;
  int* idxbuf = (int*)d_ws;   // [B][3][1000] ints = 96 KB scratch

  // Per-level top-k (per-level dynamic LDS sizing: 128KB / 32KB / 8KB)
  for (int l = 0; l < 3; ++l) {
    const int hw = 16384 >> (2 * l);
    const float* obj = (const float*)d_in[7 * l + 3];
    topk_kernel<<<NB, 1024, (size_t)hw * sizeof(unsigned long long), stream>>>(
        obj, idxbuf, l, hw);
  }

  Ptrs P;
  for (int i = 0; i < 21; ++i) P.p[i] = (const float*)d_in[i];
  gather_kernel<<<NB * 3 * NK, 256, 0, stream>>>(P, idxbuf, (float*)d_out);
}